// GTN_90228672955102
// MI455X (gfx1250) — compile-verified
//
#include <hip/hip_runtime.h>
#include <hip/hip_bf16.h>

#define DDIM 128
#define MNEI 32
#define LSTR 132   // LDS row stride: 132 mod 64 == 4 -> conflict-free A-frag reads
#define SCALEF 0.08838834764831843f

typedef float v2f __attribute__((ext_vector_type(2)));
typedef float v4f __attribute__((ext_vector_type(4)));
typedef float v8f __attribute__((ext_vector_type(8)));

// ---------------------------------------------------------------------------
// 128x128 matrix transpose (one-time, tiny): out[j,k] = in[k,j]
// ---------------------------------------------------------------------------
__global__ __launch_bounds__(256) void k_transpose128(
    const float* __restrict__ in, float* __restrict__ out)
{
    int idx = blockIdx.x * 256 + threadIdx.x;   // 0..16383
    int k = idx >> 7, j = idx & 127;
    out[(size_t)j * DDIM + k] = in[idx];
}

// ---------------------------------------------------------------------------
// fp32 WMMA GEMM: C[rows,128] = A[rows,128] @ Bt^T + rowscale*bias
// Bt is stored column-major of the effective B (i.e. Bt[j][k] = B[k][j]), so
// every B-fragment is a contiguous 8B load. 32 rows (2 M-tiles) per WG; each
// wave owns one 16-col tile and reuses each B fragment for 2 WMMAs.
// rows must be a multiple of 32. v_wmma_f32_16x16x4_f32, 32 K-steps x 2 tiles.
// ---------------------------------------------------------------------------
__global__ __launch_bounds__(256) void k_gemm128(
    const float* __restrict__ A, const float* __restrict__ Bt,
    const float* __restrict__ bias, const float* __restrict__ rowscale,
    float* __restrict__ C, int rows)
{
    __shared__ float sA[32 * LSTR];
    const int row0 = blockIdx.x * 32;
    const int tid  = threadIdx.x;
    (void)rows;
#pragma unroll
    for (int it = 0; it < 4; ++it) {
        int idx = tid + it * 256;           // 1024 float4 = 32x128
        int r   = idx >> 5;
        int c4  = (idx & 31) * 4;
        float4 v = *(const float4*)(A + (size_t)(row0 + r) * DDIM + c4);
        float* d = &sA[r * LSTR + c4];
        d[0] = v.x; d[1] = v.y; d[2] = v.z; d[3] = v.w;
    }
    __syncthreads();

    const int wave = tid >> 5, lane = tid & 31;
    const int hi = lane >> 4, ln = lane & 15;
    const int col = wave * 16 + ln;

    v8f acc0 = {}, acc1 = {};
    const float* ar0 = &sA[ln * LSTR + 2 * hi];          // tile 0: rows 0..15
    const float* ar1 = &sA[(16 + ln) * LSTR + 2 * hi];   // tile 1: rows 16..31
    const float* bp  = Bt + (size_t)col * DDIM + 2 * hi; // contiguous B frags
#pragma unroll
    for (int kb = 0; kb < DDIM; kb += 4) {
        v2f b;  b.x  = bp[kb];  b.y  = bp[kb + 1];
        v2f a0; a0.x = ar0[kb]; a0.y = ar0[kb + 1];
        v2f a1; a1.x = ar1[kb]; a1.y = ar1[kb + 1];
        acc0 = __builtin_amdgcn_wmma_f32_16x16x4_f32(false, a0, false, b,
                                                     (short)0, acc0, false, false);
        acc1 = __builtin_amdgcn_wmma_f32_16x16x4_f32(false, a1, false, b,
                                                     (short)0, acc1, false, false);
    }
    const float bc = bias ? bias[col] : 0.0f;
#pragma unroll
    for (int i = 0; i < 8; ++i) {               // C/D: VGPR i -> rows i (lo) / i+8 (hi)
        int r0 = row0 + i + 8 * hi;
        int r1 = r0 + 16;
        float s0 = rowscale ? rowscale[r0] : 1.0f;
        float s1 = rowscale ? rowscale[r1] : 1.0f;
        C[(size_t)r0 * DDIM + col] = acc0[i] + s0 * bc;
        C[(size_t)r1 * DDIM + col] = acc1[i] + s1 * bc;
    }
}

// ---------------------------------------------------------------------------
// Layer pass A: logits[n,m] = (neigh[n,m,:].u[n] + bq.key[n]) * SCALE
// One wave32 per node; lane m streams its neighbor row (NT loads, 655MB stream).
// ---------------------------------------------------------------------------
__global__ __launch_bounds__(256) void k_neigh_logits(
    const float* __restrict__ neigh, const float* __restrict__ u,
    const float* __restrict__ key, const float* __restrict__ bq,
    float* __restrict__ logits, int N)
{
    const int wave = threadIdx.x >> 5, lane = threadIdx.x & 31;
    const int n = blockIdx.x * 8 + wave;
    if (n >= N) return;

    v4f kv = *(const v4f*)(key + (size_t)n * DDIM + lane * 4);
    v4f bv = *(const v4f*)(bq + lane * 4);
    float c = kv.x * bv.x + kv.y * bv.y + kv.z * bv.z + kv.w * bv.w;
#pragma unroll
    for (int off = 16; off > 0; off >>= 1) c += __shfl_xor(c, off, 32);

    const v4f* np = (const v4f*)(neigh + ((size_t)n * MNEI + lane) * DDIM);
    const v4f* up = (const v4f*)(u + (size_t)n * DDIM);
    float acc = 0.0f;
#pragma unroll 8
    for (int d4 = 0; d4 < 32; ++d4) {
        v4f x = __builtin_nontemporal_load(np + d4);
        v4f y = up[d4];
        acc += x.x * y.x + x.y * y.y + x.z * y.z + x.w * y.w;
    }
    logits[(size_t)n * MNEI + lane] = (acc + c) * SCALEF;
}

// ---------------------------------------------------------------------------
// Global (node-axis) softmax stats: per column m -> colmax, 1/sum(exp(x-max)).
// One WG per column (M=32 WGs). Logits array (2.56MB) is L2-resident.
// ---------------------------------------------------------------------------
__global__ __launch_bounds__(256) void k_softmax_stats(
    const float* __restrict__ logits, float* __restrict__ stats, int N)
{
    __shared__ float red[256];
    const int m = blockIdx.x, tid = threadIdx.x;
    float mx = -1e30f;
    for (int n = tid; n < N; n += 256) mx = fmaxf(mx, logits[(size_t)n * MNEI + m]);
    red[tid] = mx; __syncthreads();
    for (int s = 128; s > 0; s >>= 1) {
        if (tid < s) red[tid] = fmaxf(red[tid], red[tid + s]);
        __syncthreads();
    }
    mx = red[0]; __syncthreads();
    float sum = 0.0f;
    for (int n = tid; n < N; n += 256) sum += __expf(logits[(size_t)n * MNEI + m] - mx);
    red[tid] = sum; __syncthreads();
    for (int s = 128; s > 0; s >>= 1) {
        if (tid < s) red[tid] += red[tid + s];
        __syncthreads();
    }
    if (tid == 0) { stats[m] = mx; stats[32 + m] = 1.0f / red[0]; }
}

// ---------------------------------------------------------------------------
// Layer pass C: p[n,:] = sum_m att[n,m]*neigh[n,m,:],  s[n] = sum_m att[n,m]
// One wave32 per node; lane covers features 4*lane..4*lane+3 (coalesced, NT).
// ---------------------------------------------------------------------------
__global__ __launch_bounds__(256) void k_neigh_combine(
    const float* __restrict__ neigh, const float* __restrict__ logits,
    const float* __restrict__ stats, float* __restrict__ p,
    float* __restrict__ s, int N)
{
    const int wave = threadIdx.x >> 5, lane = threadIdx.x & 31;
    const int n = blockIdx.x * 8 + wave;
    if (n >= N) return;

    float a = __expf(logits[(size_t)n * MNEI + lane] - stats[lane]) * stats[32 + lane];
    float ssum = a;
#pragma unroll
    for (int off = 16; off > 0; off >>= 1) ssum += __shfl_xor(ssum, off, 32);

    v4f acc = {0.f, 0.f, 0.f, 0.f};
    const v4f* base = (const v4f*)(neigh + (size_t)n * MNEI * DDIM);
#pragma unroll 8
    for (int m = 0; m < MNEI; ++m) {
        float am = __shfl(a, m, 32);
        v4f x = __builtin_nontemporal_load(base + m * 32 + lane);
        acc.x += am * x.x; acc.y += am * x.y; acc.z += am * x.z; acc.w += am * x.w;
    }
    *(v4f*)(p + (size_t)n * DDIM + lane * 4) = acc;
    if (lane == 0) s[n] = ssum;
}

// ---------------------------------------------------------------------------
// Pack H[3N,128] = interleave(tgt, h0, h1) per node.
// ---------------------------------------------------------------------------
__global__ __launch_bounds__(256) void k_pack_h(
    const float* __restrict__ tgt, const float* __restrict__ h0,
    const float* __restrict__ h1, float* __restrict__ Hf, int N)
{
    int idx = blockIdx.x * 256 + threadIdx.x;       // float4 index over 3N*32
    if (idx >= N * 3 * 32) return;
    int g = idx >> 5, c = idx & 31;
    int node = g / 3, slot = g - node * 3;
    const float* src = (slot == 0) ? tgt : ((slot == 1) ? h0 : h1);
    ((float4*)Hf)[idx] = ((const float4*)(src + (size_t)node * DDIM))[c];
}

// ---------------------------------------------------------------------------
// Final: per-node 3x3 attention + residual + LayerNorm. One wave32 per node.
// ---------------------------------------------------------------------------
__global__ __launch_bounds__(256) void k_attn_ln(
    const float* __restrict__ Qf, const float* __restrict__ Kf,
    const float* __restrict__ Vf, const float* __restrict__ Rf,
    const float* __restrict__ gamma, const float* __restrict__ beta,
    float* __restrict__ out, int N)
{
    const int wave = threadIdx.x >> 5, lane = threadIdx.x & 31;
    const int n = blockIdx.x * 8 + wave;
    if (n >= N) return;
    const size_t base = (size_t)n * 3 * DDIM;

    float q[3][4], k[3][4];
#pragma unroll
    for (int i = 0; i < 3; ++i)
#pragma unroll
        for (int t = 0; t < 4; ++t) {
            q[i][t] = Qf[base + i * DDIM + lane + 32 * t];
            k[i][t] = Kf[base + i * DDIM + lane + 32 * t];
        }
    float att[3][3];
#pragma unroll
    for (int i = 0; i < 3; ++i)
#pragma unroll
        for (int j = 0; j < 3; ++j) {
            float part = q[i][0]*k[j][0] + q[i][1]*k[j][1] + q[i][2]*k[j][2] + q[i][3]*k[j][3];
#pragma unroll
            for (int off = 16; off > 0; off >>= 1) part += __shfl_xor(part, off, 32);
            att[i][j] = part * SCALEF;
        }
#pragma unroll
    for (int i = 0; i < 3; ++i) {
        float mx = fmaxf(att[i][0], fmaxf(att[i][1], att[i][2]));
        float e0 = __expf(att[i][0] - mx), e1 = __expf(att[i][1] - mx), e2 = __expf(att[i][2] - mx);
        float r = 1.0f / (e0 + e1 + e2);
        att[i][0] = e0 * r; att[i][1] = e1 * r; att[i][2] = e2 * r;
    }
    v4f v0 = *(const v4f*)(Vf + base + lane * 4);
    v4f v1 = *(const v4f*)(Vf + base + DDIM + lane * 4);
    v4f v2 = *(const v4f*)(Vf + base + 2 * DDIM + lane * 4);
    v4f g4 = *(const v4f*)(gamma + lane * 4);
    v4f b4 = *(const v4f*)(beta + lane * 4);
#pragma unroll
    for (int i = 0; i < 3; ++i) {
        v4f r4 = *(const v4f*)(Rf + base + i * DDIM + lane * 4);
        v4f o;
        o.x = att[i][0]*v0.x + att[i][1]*v1.x + att[i][2]*v2.x + r4.x;
        o.y = att[i][0]*v0.y + att[i][1]*v1.y + att[i][2]*v2.y + r4.y;
        o.z = att[i][0]*v0.z + att[i][1]*v1.z + att[i][2]*v2.z + r4.z;
        o.w = att[i][0]*v0.w + att[i][1]*v1.w + att[i][2]*v2.w + r4.w;
        float msum = o.x + o.y + o.z + o.w;
#pragma unroll
        for (int off = 16; off > 0; off >>= 1) msum += __shfl_xor(msum, off, 32);
        float mu = msum * (1.0f / 128.0f);
        float dx = o.x - mu, dy = o.y - mu, dz = o.z - mu, dw = o.w - mu;
        float vsum = dx*dx + dy*dy + dz*dz + dw*dw;
#pragma unroll
        for (int off = 16; off > 0; off >>= 1) vsum += __shfl_xor(vsum, off, 32);
        float rstd = rsqrtf(vsum * (1.0f / 128.0f) + 1e-5f);
        v4f r;
        r.x = dx * rstd * g4.x + b4.x;
        r.y = dy * rstd * g4.y + b4.y;
        r.z = dz * rstd * g4.z + b4.z;
        r.w = dw * rstd * g4.w + b4.w;
        __builtin_nontemporal_store(r, (v4f*)(out + base + i * DDIM + lane * 4));
    }
}

// ---------------------------------------------------------------------------
extern "C" void kernel_launch(void* const* d_in, const int* in_sizes, int n_in,
                              void* d_out, int out_size, void* d_ws, size_t ws_size,
                              hipStream_t stream)
{
    (void)n_in; (void)out_size; (void)ws_size;
    const float* tgt = (const float*)d_in[0];
    const float* neigh[2] = {(const float*)d_in[1], (const float*)d_in[2]};
    const int N = in_sizes[0] / DDIM;
    const size_t ND = (size_t)N * DDIM;
    const size_t WSZ = (size_t)DDIM * DDIM;   // 16384 floats per weight matrix

    float* w  = (float*)d_ws;
    float* h0 = w;                    // [N,128] persists
    float* h1 = w + ND;               // [N,128] persists
    float* Hf = w + 2 * ND;           // [3N,128]
    float* Qf = Hf + 3 * ND;
    float* Kf = Qf + 3 * ND;
    float* Vf = Kf + 3 * ND;
    float* Rf = Vf + 3 * ND;          // ends at 17*ND
    float* tw = Rf + 3 * ND;          // 8 transposed weight matrices (0.5 MB)
    // layer scratch aliases the Hf.. region (dead by the time Hf/Qf are written)
    float* key    = Hf;
    float* u      = key + ND;
    float* logits = u + ND;
    float* stats  = logits + (size_t)N * MNEI;
    float* p      = stats + 64;
    float* s      = p + ND;

    dim3 blk(256);
    const int gN32 = N / 32;          // 32-row GEMM tiles
    const int gN8  = N / 8;           // 8 nodes (waves) per block

    // One-time weight transposes so every GEMM uses contiguous B fragments.
    // tw slots: 0:l0_Wk^T 1:l0_Wv^T 2:l1_Wk^T 3:l1_Wv^T 4:Wq^T 5:Wk^T 6:Wv^T 7:Wres^T
    const int tsrc[8] = {3, 7, 9, 13, 17, 15, 19, 21};
    for (int i = 0; i < 8; ++i)
        k_transpose128<<<64, blk, 0, stream>>>((const float*)d_in[tsrc[i]], tw + i * WSZ);

    float* hbuf[2] = {h0, h1};
    for (int li = 0; li < 2; ++li) {
        const float* bk = (const float*)d_in[3 + li * 6 + 1];
        const float* Wq = (const float*)d_in[3 + li * 6 + 2];
        const float* bq = (const float*)d_in[3 + li * 6 + 3];
        const float* bv = (const float*)d_in[3 + li * 6 + 5];
        const float* WkT = tw + (li * 2 + 0) * WSZ;
        const float* WvT = tw + (li * 2 + 1) * WSZ;
        // key = tgt @ Wk + bk
        k_gemm128<<<gN32, blk, 0, stream>>>(tgt, WkT, bk, nullptr, key, N);
        // u = key @ Wq^T  (effective B = Wq^T, so "Bt" = original Wq)
        k_gemm128<<<gN32, blk, 0, stream>>>(key, Wq, nullptr, nullptr, u, N);
        k_neigh_logits<<<gN8, blk, 0, stream>>>(neigh[li], u, key, bq, logits, N);
        k_softmax_stats<<<MNEI, blk, 0, stream>>>(logits, stats, N);
        k_neigh_combine<<<gN8, blk, 0, stream>>>(neigh[li], logits, stats, p, s, N);
        // h_l = p @ Wv + s*bv
        k_gemm128<<<gN32, blk, 0, stream>>>(p, WvT, bv, s, hbuf[li], N);
    }

    const float* bko  = (const float*)d_in[16];
    const float* bqo  = (const float*)d_in[18];
    const float* bvo  = (const float*)d_in[20];
    const float* bres = (const float*)d_in[22];
    const float* gam  = (const float*)d_in[23];
    const float* bet  = (const float*)d_in[24];

    k_pack_h<<<(N * 3 * 32 + 255) / 256, blk, 0, stream>>>(tgt, h0, h1, Hf, N);
    const int g3N32 = (3 * N) / 32;
    k_gemm128<<<g3N32, blk, 0, stream>>>(Hf, tw + 4 * WSZ, bqo,  nullptr, Qf, 3 * N);
    k_gemm128<<<g3N32, blk, 0, stream>>>(Hf, tw + 5 * WSZ, bko,  nullptr, Kf, 3 * N);
    k_gemm128<<<g3N32, blk, 0, stream>>>(Hf, tw + 6 * WSZ, bvo,  nullptr, Vf, 3 * N);
    k_gemm128<<<g3N32, blk, 0, stream>>>(Hf, tw + 7 * WSZ, bres, nullptr, Rf, 3 * N);
    k_attn_ln<<<gN8, blk, 0, stream>>>(Qf, Kf, Vf, Rf, gam, bet, (float*)d_out, N);
}